// GATClassifier_26731876450813
// MI455X (gfx1250) — compile-verified
//
#include <hip/hip_runtime.h>
#include <math.h>

typedef float v2f __attribute__((ext_vector_type(2)));
typedef float v8f __attribute__((ext_vector_type(8)));

// ---------- monotonic float<->uint key for atomicMax on signed floats ----------
__device__ __forceinline__ unsigned fkey(float f) {
    unsigned u = __float_as_uint(f);
    return (u & 0x80000000u) ? ~u : (u | 0x80000000u);
}
__device__ __forceinline__ float funkey(unsigned k) {
    return (k & 0x80000000u) ? __uint_as_float(k & 0x7fffffffu)
                             : __uint_as_float(~k);
}
// key of -inf (init value for segment max)
#define NEG_INF_KEY 0x007FFFFFu

// ---------- fills ----------
__global__ void fill_f32(float* __restrict__ p, float v, long long n) {
    long long i = (long long)blockIdx.x * blockDim.x + threadIdx.x;
    if (i < n) p[i] = v;
}
__global__ void fill_u32(unsigned* __restrict__ p, unsigned v, long long n) {
    long long i = (long long)blockIdx.x * blockDim.x + threadIdx.x;
    if (i < n) p[i] = v;
}

// ---------- WMMA fp32 GEMM: C[M,Ncols] = A[M,K] @ B[K,Ncols] (+ bias[col]) ----------
// One block per 16-row tile of A (staged in LDS); NTILES waves, one 16-col tile each.
// Uses V_WMMA_F32_16X16X4_F32: A 16x4 (2 VGPR/lane), B 4x16 (2 VGPR/lane), C 16x16 (8 VGPR/lane).
template <int K, int NTILES>
__global__ __launch_bounds__(32 * NTILES)
void gemm_wmma(const float* __restrict__ A, const float* __restrict__ B,
               const float* __restrict__ bias, float* __restrict__ C, int Ncols) {
    __shared__ float As[16][K + 4];   // +4 pad: conflict-free column reads
    const int mTile = blockIdx.x;
    const int tid   = threadIdx.x;

    // cooperative, coalesced load of the 16xK A tile
    const float* Ablk = A + (size_t)mTile * 16 * K;
    for (int i = tid; i < 16 * K; i += 32 * NTILES) {
        As[i / K][i % K] = Ablk[i];
    }
    __syncthreads();

    const int lane  = tid & 31;
    const int nTile = tid >> 5;
    const int col   = nTile * 16 + (lane & 15);
    const int kOff  = (lane >> 4) << 1;   // lanes 0-15: K=k,k+1 ; lanes 16-31: K=k+2,k+3
    const int rowA  = lane & 15;

    v8f acc = {};
    for (int k = 0; k < K; k += 4) {
        v2f a, b;
        a.x = As[rowA][k + kOff];
        a.y = As[rowA][k + kOff + 1];
        b.x = B[(size_t)(k + kOff)     * Ncols + col];
        b.y = B[(size_t)(k + kOff + 1) * Ncols + col];
        acc = __builtin_amdgcn_wmma_f32_16x16x4_f32(false, a, false, b,
                                                    (short)0, acc, false, false);
    }

    // C/D layout: VGPR j -> row j (lanes 0-15) / row j+8 (lanes 16-31), col = lane&15
    const int rowBase = mTile * 16 + ((lane >> 4) << 3);
    const float bc = bias ? bias[col] : 0.0f;
#pragma unroll
    for (int j = 0; j < 8; ++j) {
        C[(size_t)(rowBase + j) * Ncols + col] = acc[j] + bc;
    }
}

// ---------- attention scores: el[n,h] = <feat[n,h,:], al[h,:]>, er likewise ----------
template <int D>
__global__ void attn_scores(const float* __restrict__ feat, const float* __restrict__ al,
                            const float* __restrict__ ar, float* __restrict__ el,
                            float* __restrict__ er, int NH) {
    int i = blockIdx.x * blockDim.x + threadIdx.x;
    if (i >= NH) return;
    int h = i & 3;                          // H == 4
    const float* f = feat + (size_t)i * D;
    float sl = 0.f, sr = 0.f;
#pragma unroll
    for (int k = 0; k < D; ++k) {
        float v = f[k];
        sl += v * al[h * D + k];
        sr += v * ar[h * D + k];
    }
    el[i] = sl;
    er[i] = sr;
}

__device__ __forceinline__ float leaky(float x) { return x > 0.f ? x : 0.2f * x; }

// ---------- edge pass 1: segment max (via uint-key atomicMax) ----------
__global__ void edge_max(const int* __restrict__ src, const int* __restrict__ dst,
                         const float* __restrict__ el, const float* __restrict__ er,
                         unsigned* __restrict__ mkey, int EH) {
    int i = blockIdx.x * blockDim.x + threadIdx.x;
    if (i >= EH) return;
    int e = i >> 2, h = i & 3;
    int s = src[e], d = dst[e];
    float x = leaky(el[s * 4 + h] + er[d * 4 + h]);
    atomicMax(&mkey[d * 4 + h], fkey(x));
}

// ---------- edge pass 2: segment sum of exp(e - max) ----------
__global__ void edge_expsum(const int* __restrict__ src, const int* __restrict__ dst,
                            const float* __restrict__ el, const float* __restrict__ er,
                            const unsigned* __restrict__ mkey, float* __restrict__ z, int EH) {
    int i = blockIdx.x * blockDim.x + threadIdx.x;
    if (i >= EH) return;
    int e = i >> 2, h = i & 3;
    int s = src[e], d = dst[e];
    int dh = d * 4 + h;
    float x = leaky(el[s * 4 + h] + er[dh]);
    atomicAdd(&z[dh], expf(x - funkey(mkey[dh])));
}

// ---------- edge pass 3: agg[dst,h,:] += alpha * feat[src,h,:] ----------
template <int D>
__global__ void edge_agg(const int* __restrict__ src, const int* __restrict__ dst,
                         const float* __restrict__ el, const float* __restrict__ er,
                         const unsigned* __restrict__ mkey, const float* __restrict__ z,
                         const float* __restrict__ feat, float* __restrict__ agg, int EH) {
    int i = blockIdx.x * blockDim.x + threadIdx.x;
    if (i >= EH) return;
    int e = i >> 2, h = i & 3;
    int s = src[e], d = dst[e];
    int dh = d * 4 + h;
    float x = leaky(el[s * 4 + h] + er[dh]);
    float alpha = expf(x - funkey(mkey[dh])) / z[dh];
    const float* fs = feat + ((size_t)s * 4 + h) * D;
    float* ad       = agg  + ((size_t)d * 4 + h) * D;
#pragma unroll
    for (int k = 0; k < D; ++k) atomicAdd(&ad[k], fs[k] * alpha);
}

// ---------- bias + optional ReLU, in place ----------
template <int BD, bool RELU>
__global__ void bias_act(float* __restrict__ x, const float* __restrict__ bias, long long n) {
    long long i = (long long)blockIdx.x * blockDim.x + threadIdx.x;
    if (i >= n) return;
    float v = x[i] + bias[i & (BD - 1)];
    if (RELU) v = fmaxf(v, 0.f);
    x[i] = v;
}

// ---------- finalize: head mean + bias + residual, then log_softmax ----------
__global__ void final_kernel(const float* __restrict__ agg2, const float* __restrict__ b2,
                             const float* __restrict__ res, float* __restrict__ out_h,
                             float* __restrict__ out_ls, float* __restrict__ out_scalar, int N) {
    int n = blockIdx.x * blockDim.x + threadIdx.x;
    if (n == 0) out_scalar[0] = 0.0f;
    if (n >= N) return;
    float hv[16];
#pragma unroll
    for (int c = 0; c < 16; ++c) {
        float acc = 0.f;
#pragma unroll
        for (int h = 0; h < 4; ++h) acc += agg2[(size_t)n * 64 + h * 16 + c] + b2[h * 16 + c];
        hv[c] = 0.25f * acc + res[(size_t)n * 16 + c];   // res already includes bres
    }
    float mx = hv[0];
#pragma unroll
    for (int c = 1; c < 16; ++c) mx = fmaxf(mx, hv[c]);
    float s = 0.f;
#pragma unroll
    for (int c = 0; c < 16; ++c) s += expf(hv[c] - mx);
    float lse = mx + logf(s);
#pragma unroll
    for (int c = 0; c < 16; ++c) {
        out_h[(size_t)n * 16 + c]  = hv[c];
        out_ls[(size_t)n * 16 + c] = hv[c] - lse;
    }
}

static inline int cdiv(long long a, int b) { return (int)((a + b - 1) / b); }

extern "C" void kernel_launch(void* const* d_in, const int* in_sizes, int n_in,
                              void* d_out, int out_size, void* d_ws, size_t ws_size,
                              hipStream_t stream) {
    const float* X    = (const float*)d_in[0];
    const float* W1   = (const float*)d_in[1];
    const float* al1  = (const float*)d_in[2];
    const float* ar1  = (const float*)d_in[3];
    const float* b1   = (const float*)d_in[4];
    const float* W2   = (const float*)d_in[5];
    const float* al2  = (const float*)d_in[6];
    const float* ar2  = (const float*)d_in[7];
    const float* b2   = (const float*)d_in[8];
    const float* Wres = (const float*)d_in[9];
    const float* bres = (const float*)d_in[10];
    const int*   src  = (const int*)d_in[11];
    const int*   dst  = (const int*)d_in[12];

    const int N = in_sizes[0] / 256;   // 50000 (multiple of 16)
    const int E = in_sizes[11];        // 800000

    // workspace layout (floats). feat region reused by layer 2 (feat2 + agg2).
    float*    ws   = (float*)d_ws;
    float*    feat = ws;                                   // L1: feat1 [N*128]; L2: feat2 [N*64] @0, agg2 [N*64] @N*64
    float*    h1   = ws + (size_t)N * 128;                 // agg1 -> h1 in place [N*128]
    float*    el   = ws + (size_t)N * 256;                 // [N*4]
    float*    er   = el + (size_t)N * 4;                   // [N*4]
    unsigned* mkey = (unsigned*)(er + (size_t)N * 4);      // [N*4]
    float*    z    = (float*)mkey + (size_t)N * 4;         // [N*4]
    float*    res  = z + (size_t)N * 4;                    // [N*16]

    const int mt = N / 16;   // 3125 row tiles

    // ===================== layer 1 =====================
    gemm_wmma<256, 8><<<mt, 256, 0, stream>>>(X, W1, nullptr, feat, 128);
    attn_scores<32><<<cdiv((long long)N * 4, 256), 256, 0, stream>>>(feat, al1, ar1, el, er, N * 4);
    fill_u32<<<cdiv((long long)N * 4, 256), 256, 0, stream>>>(mkey, NEG_INF_KEY, (long long)N * 4);
    fill_f32<<<cdiv((long long)N * 4, 256), 256, 0, stream>>>(z, 0.f, (long long)N * 4);
    fill_f32<<<cdiv((long long)N * 128, 256), 256, 0, stream>>>(h1, 0.f, (long long)N * 128);
    edge_max<<<cdiv((long long)E * 4, 256), 256, 0, stream>>>(src, dst, el, er, mkey, E * 4);
    edge_expsum<<<cdiv((long long)E * 4, 256), 256, 0, stream>>>(src, dst, el, er, mkey, z, E * 4);
    edge_agg<32><<<cdiv((long long)E * 4, 256), 256, 0, stream>>>(src, dst, el, er, mkey, z, feat, h1, E * 4);
    bias_act<128, true><<<cdiv((long long)N * 128, 256), 256, 0, stream>>>(h1, b1, (long long)N * 128);

    // ===================== layer 2 =====================
    float* feat2 = feat;                       // [N*64], overwrites feat1 (no longer needed)
    float* agg2  = feat + (size_t)N * 64;      // [N*64]
    gemm_wmma<128, 4><<<mt, 128, 0, stream>>>(h1, W2, nullptr, feat2, 64);
    attn_scores<16><<<cdiv((long long)N * 4, 256), 256, 0, stream>>>(feat2, al2, ar2, el, er, N * 4);
    fill_u32<<<cdiv((long long)N * 4, 256), 256, 0, stream>>>(mkey, NEG_INF_KEY, (long long)N * 4);
    fill_f32<<<cdiv((long long)N * 4, 256), 256, 0, stream>>>(z, 0.f, (long long)N * 4);
    fill_f32<<<cdiv((long long)N * 64, 256), 256, 0, stream>>>(agg2, 0.f, (long long)N * 64);
    edge_max<<<cdiv((long long)E * 4, 256), 256, 0, stream>>>(src, dst, el, er, mkey, E * 4);
    edge_expsum<<<cdiv((long long)E * 4, 256), 256, 0, stream>>>(src, dst, el, er, mkey, z, E * 4);
    edge_agg<16><<<cdiv((long long)E * 4, 256), 256, 0, stream>>>(src, dst, el, er, mkey, z, feat2, agg2, E * 4);

    // ===================== residual + finalize =====================
    gemm_wmma<256, 1><<<mt, 32, 0, stream>>>(X, Wres, bres, res, 16);

    float* out = (float*)d_out;
    final_kernel<<<cdiv(N, 256), 256, 0, stream>>>(agg2, b2, res, out,
                                                   out + (size_t)N * 16 + 1,   // log_softmax after scalar
                                                   out + (size_t)N * 16,       // scalar 0.0
                                                   N);
}